// TPUGraphNetwork_78091095376379
// MI455X (gfx1250) — compile-verified
//
#include <hip/hip_runtime.h>
#include <math.h>

typedef __attribute__((ext_vector_type(2))) float v2f;
typedef __attribute__((ext_vector_type(8))) float v8f;

#define GNN_D 128
#define GNN_NUM_GRAPHS 10
#define CSR_CHUNK 1024

// ---------------------------------------------------------------- zero fill
static __global__ void fill_zero_u32(unsigned int* __restrict__ p, long long n) {
    long long i = (long long)blockIdx.x * blockDim.x + threadIdx.x;
    long long stride = (long long)gridDim.x * blockDim.x;
    for (; i < n; i += stride) p[i] = 0u;
}

// ------------------------------------------------- embedding + concat layer
// H[n, 0:32]  = op_emb[int(feat[n,0]), :] ; H[n, 32:128] = feat[n, 1:97]
static __global__ void embed_kernel(const float* __restrict__ feat,
                                    const float* __restrict__ op_emb,
                                    float* __restrict__ H, int Nn, int F) {
    long long idx = (long long)blockIdx.x * blockDim.x + threadIdx.x;
    if (idx >= (long long)Nn * GNN_D) return;
    int n = (int)(idx >> 7);
    int d = (int)(idx & (GNN_D - 1));
    const float* f = feat + (long long)n * F;
    float v;
    if (d < 32) {
        int op = (int)f[0];
        v = op_emb[op * 32 + d];
    } else {
        v = f[d - 31];
    }
    H[idx] = v;
}

// ------------------------------------------------------------ CSR building
static __global__ void csr_count_kernel(const int* __restrict__ rows,
                                        int* __restrict__ counts, int E) {
    int e = blockIdx.x * blockDim.x + threadIdx.x;
    if (e < E) atomicAdd(&counts[rows[e]], 1);
}

static __global__ void csr_block_reduce(const int* __restrict__ counts,
                                        int* __restrict__ blocksum, int Nn) {
    __shared__ int s[256];
    int base = blockIdx.x * CSR_CHUNK;
    int t = 0;
    for (int i = threadIdx.x; i < CSR_CHUNK; i += 256) {
        int gi = base + i;
        if (gi < Nn) t += counts[gi];
    }
    s[threadIdx.x] = t;
    __syncthreads();
    for (int off = 128; off > 0; off >>= 1) {
        if ((int)threadIdx.x < off) s[threadIdx.x] += s[threadIdx.x + off];
        __syncthreads();
    }
    if (threadIdx.x == 0) blocksum[blockIdx.x] = s[0];
}

static __global__ void csr_scan_blocksum(int* __restrict__ blocksum, int nb,
                                         int* __restrict__ rowstart, int Nn) {
    if (threadIdx.x == 0 && blockIdx.x == 0) {
        int acc = 0;
        for (int b = 0; b < nb; ++b) { int t = blocksum[b]; blocksum[b] = acc; acc += t; }
        rowstart[Nn] = acc;                      // == E
    }
}

static __global__ void csr_scan_write(const int* __restrict__ counts,
                                      const int* __restrict__ blocksum,
                                      int* __restrict__ rowstart,
                                      int* __restrict__ cursor, int Nn) {
    __shared__ int s[256];
    int base = blockIdx.x * CSR_CHUNK + threadIdx.x * 4;
    int c0 = 0, c1 = 0, c2 = 0, c3 = 0;
    if (base + 0 < Nn) c0 = counts[base + 0];
    if (base + 1 < Nn) c1 = counts[base + 1];
    if (base + 2 < Nn) c2 = counts[base + 2];
    if (base + 3 < Nn) c3 = counts[base + 3];
    int tot = c0 + c1 + c2 + c3;
    s[threadIdx.x] = tot;
    __syncthreads();
    for (int off = 1; off < 256; off <<= 1) {     // Hillis-Steele inclusive
        int v = ((int)threadIdx.x >= off) ? s[threadIdx.x - off] : 0;
        __syncthreads();
        s[threadIdx.x] += v;
        __syncthreads();
    }
    int prefix = blocksum[blockIdx.x] + s[threadIdx.x] - tot;  // exclusive
    int p0 = prefix, p1 = p0 + c0, p2 = p1 + c1, p3 = p2 + c2;
    if (base + 0 < Nn) { rowstart[base + 0] = p0; cursor[base + 0] = p0; }
    if (base + 1 < Nn) { rowstart[base + 1] = p1; cursor[base + 1] = p1; }
    if (base + 2 < Nn) { rowstart[base + 2] = p2; cursor[base + 2] = p2; }
    if (base + 3 < Nn) { rowstart[base + 3] = p3; cursor[base + 3] = p3; }
}

static __global__ void csr_scatter_kernel(const int* __restrict__ rows,
                                          const int* __restrict__ cols,
                                          const int* __restrict__ codes,
                                          int* __restrict__ cursor,
                                          int* __restrict__ ecol,
                                          float* __restrict__ eval, int E) {
    int e = blockIdx.x * blockDim.x + threadIdx.x;
    if (e >= E) return;
    int idx = atomicAdd(&cursor[rows[e]], 1);
    ecol[idx] = cols[e];
    eval[idx] = ldexpf(1.0f, -(codes[e] & 127));   // DECAY^(code % 128), exact
}

// -------------------------------------------------------- Z = H @ W + bias
// One wave computes a 16x128 slab of Z using v_wmma_f32_16x16x4_f32.
// W is staged in LDS pair-interleaved so each B operand is one ds_load_b64.
static __global__ void gemm_bias_kernel(const float* __restrict__ H,
                                        const float* __restrict__ W,
                                        const float* __restrict__ bias,
                                        float* __restrict__ Z,
                                        int Nn, int ntiles) {
    __shared__ float Wl[GNN_D * GNN_D];   // [(k>>1)*128 + col][2] = W[2t+j][col]
    for (int i = threadIdx.x; i < GNN_D * GNN_D; i += blockDim.x) {
        int k = i >> 7, col = i & 127;
        Wl[((((k >> 1) << 7) + col) << 1) | (k & 1)] = W[i];
    }
    __syncthreads();

    int wave = (int)((blockIdx.x * (unsigned)blockDim.x + threadIdx.x) >> 5);
    if (wave >= ntiles) return;           // wave-uniform: EXEC stays all-ones
    int lane = threadIdx.x & 31;
    int m  = lane & 15;
    int hl = lane >> 4;
    int row0 = wave << 4;

    int arow = row0 + m;
    if (arow >= Nn) arow = Nn - 1;        // clamp (N is a multiple of 16 here)
    const float* Arow = H + (long long)arow * GNN_D;

    v8f acc[8] = {};
    for (int kk = 0; kk < 32; ++kk) {
        int k0 = kk * 4 + hl * 2;
        v2f a = *(const v2f*)(Arow + k0);              // global b64
        const float* Wt = Wl + (((kk * 2 + hl) << 7) << 1);
#pragma unroll
        for (int j = 0; j < 8; ++j) {
            int col = j * 16 + m;
            v2f b = *(const v2f*)(Wt + (col << 1));    // ds b64
            acc[j] = __builtin_amdgcn_wmma_f32_16x16x4_f32(
                false, a, false, b, (short)0, acc[j], false, false);
        }
    }

    if (row0 + 16 <= Nn) {                // wave-uniform fast path
#pragma unroll
        for (int j = 0; j < 8; ++j) {
            int col = j * 16 + m;
            float bv = bias[col];
            float* Zc = Z + (long long)(row0 + hl * 8) * GNN_D + col;
#pragma unroll
            for (int v = 0; v < 8; ++v)
                Zc[(long long)v * GNN_D] = acc[j][v] + bv;
        }
    } else {
#pragma unroll
        for (int j = 0; j < 8; ++j) {
            int col = j * 16 + m;
            float bv = bias[col];
#pragma unroll
            for (int v = 0; v < 8; ++v) {
                int r = row0 + v + hl * 8;
                if (r < Nn) Z[(long long)r * GNN_D + col] = acc[j][v] + bv;
            }
        }
    }
}

// ------------------------- H[r] = relu( sum_{e in row r} ev[e] * Z[col[e]] )
// Atomic-free CSR SpMM: one wave per row, lane owns 4 columns, b128 gathers
// resolve in the 192MB L2 (whole working set is L2-resident on MI455X).
static __global__ void spmm_csr_kernel(const float* __restrict__ Z,
                                       float* __restrict__ Hout,
                                       const int* __restrict__ rowstart,
                                       const int* __restrict__ ecol,
                                       const float* __restrict__ eval, int Nn) {
    int r = (int)((blockIdx.x * (unsigned)blockDim.x + threadIdx.x) >> 5);
    if (r >= Nn) return;
    int lane = threadIdx.x & 31;
    int s = rowstart[r], e = rowstart[r + 1];
    float4 acc = make_float4(0.f, 0.f, 0.f, 0.f);
    for (int i = s; i < e; ++i) {
        int c = ecol[i];
        float ev = eval[i];
        const float4* src = (const float4*)(Z + (long long)c * GNN_D);
        float4 v = src[lane];
        acc.x += ev * v.x; acc.y += ev * v.y;
        acc.z += ev * v.z; acc.w += ev * v.w;
    }
    float4 o;                                    // fused ReLU
    o.x = fmaxf(acc.x, 0.f); o.y = fmaxf(acc.y, 0.f);
    o.z = fmaxf(acc.z, 0.f); o.w = fmaxf(acc.w, 0.f);
    ((float4*)(Hout + (long long)r * GNN_D))[lane] = o;
}

// --------------------------- out[g, cfg] += | H[n,:] . Wp + bp |  (per node)
static __global__ void proj_kernel(const float* __restrict__ H,
                                   const float* __restrict__ Wp,
                                   const float* __restrict__ bp,
                                   const int* __restrict__ gidx,
                                   float* __restrict__ out,
                                   int Nn, int cfg, int C) {
    int n = (int)((blockIdx.x * (unsigned)blockDim.x + threadIdx.x) >> 5);
    if (n >= Nn) return;
    int lane = threadIdx.x & 31;
    const float* h = H + (long long)n * GNN_D + lane * 4;
    const float* w = Wp + lane * 4;
    float s = h[0] * w[0] + h[1] * w[1] + h[2] * w[2] + h[3] * w[3];
#pragma unroll
    for (int off = 16; off > 0; off >>= 1) s += __shfl_down(s, off, 32);
    if (lane == 0) {
        float r = fabsf(s + bp[0]);
        unsafeAtomicAdd(&out[gidx[n] * C + cfg], r);
    }
}

// ---------------------------------------------------------------- dispatch
extern "C" void kernel_launch(void* const* d_in, const int* in_sizes, int n_in,
                              void* d_out, int out_size, void* d_ws, size_t ws_size,
                              hipStream_t stream) {
    const float* features = (const float*)d_in[0];
    const float* op_emb   = (const float*)d_in[1];
    const float* W1 = (const float*)d_in[2];
    const float* b1 = (const float*)d_in[3];
    const float* W2 = (const float*)d_in[4];
    const float* b2 = (const float*)d_in[5];
    const float* Wp = (const float*)d_in[6];
    const float* bp = (const float*)d_in[7];
    const int* rows  = (const int*)d_in[8];
    const int* cols  = (const int*)d_in[9];
    const int* codes = (const int*)d_in[10];
    const int* gidx  = (const int*)d_in[11];

    const int N = in_sizes[11];                       // graph_index is (N,)
    const int E = in_sizes[8];                        // row_indices is (E,)
    const int C = out_size / GNN_NUM_GRAPHS;          // output is (G, C)
    const int F = (int)((long long)in_sizes[0] / ((long long)C * N));
    const long long ND = (long long)N * GNN_D;
    const int nb = (N + CSR_CHUNK - 1) / CSR_CHUNK;

    // workspace layout
    float* B0      = (float*)d_ws;                    // N x 128
    float* B1      = B0 + ND;                         // N x 128
    float* eval    = B1 + ND;                         // E
    int*   ecol    = (int*)(eval + E);                // E
    int*   counts  = ecol + E;                        // N
    int*   cursor  = counts + N;                      // N
    int*   rowstart= cursor + N;                      // N + 1
    int*   blocksum= rowstart + N + 1;                // nb

    const dim3 blk(256);
    const int eblocks   = (E + 255) / 256;
    const int ntiles    = (N + 15) / 16;
    const int gemm_blocks = (ntiles + 7) / 8;
    const int wavN_blocks = (N + 7) / 8;
    const int elem_blocks = (int)((ND + 255) / 256);

    // ---- build CSR once (shared by all 4 SpMM passes) ----
    fill_zero_u32<<<(N + 255) / 256, blk, 0, stream>>>((unsigned int*)counts, N);
    csr_count_kernel<<<eblocks, blk, 0, stream>>>(rows, counts, E);
    csr_block_reduce<<<nb, blk, 0, stream>>>(counts, blocksum, N);
    csr_scan_blocksum<<<1, 32, 0, stream>>>(blocksum, nb, rowstart, N);
    csr_scan_write<<<nb, blk, 0, stream>>>(counts, blocksum, rowstart, cursor, N);
    csr_scatter_kernel<<<eblocks, blk, 0, stream>>>(rows, cols, codes, cursor, ecol, eval, E);

    fill_zero_u32<<<1, 64, 0, stream>>>((unsigned int*)d_out, out_size);

    for (int c = 0; c < C; ++c) {
        const float* feat_c = features + (long long)c * N * F;
        embed_kernel<<<elem_blocks, blk, 0, stream>>>(feat_c, op_emb, B0, N, F);
        // layer 1
        gemm_bias_kernel<<<gemm_blocks, blk, 0, stream>>>(B0, W1, b1, B1, N, ntiles);
        spmm_csr_kernel<<<wavN_blocks, blk, 0, stream>>>(B1, B0, rowstart, ecol, eval, N);
        // layer 2
        gemm_bias_kernel<<<gemm_blocks, blk, 0, stream>>>(B0, W2, b2, B1, N, ntiles);
        spmm_csr_kernel<<<wavN_blocks, blk, 0, stream>>>(B1, B0, rowstart, ecol, eval, N);
        // projection + per-graph segment sum
        proj_kernel<<<wavN_blocks, blk, 0, stream>>>(B0, Wp, bp, gidx, (float*)d_out, N, c, C);
    }
}